// EMOGI_45878840656370
// MI455X (gfx1250) — compile-verified
//
#include <hip/hip_runtime.h>
#include <hip/hip_bf16.h>
#include <math.h>

typedef __attribute__((ext_vector_type(2))) float v2f;
typedef __attribute__((ext_vector_type(8))) float v8f;

#define D 128  // D_IN == HID == 128

// ---------------- fill ----------------
__global__ void fill_kernel(float* __restrict__ p, float v, int n) {
    int i = blockIdx.x * blockDim.x + threadIdx.x;
    if (i < n) p[i] = v;
}

__global__ void fill4_kernel(float4* __restrict__ p, float v, int n4) {
    int i = blockIdx.x * blockDim.x + threadIdx.x;
    if (i < n4) p[i] = make_float4(v, v, v, v);
}

// ---------------- degree + rsqrt ----------------
__global__ void degree_kernel(const int* __restrict__ col, float* __restrict__ deg, int nE) {
    int i = blockIdx.x * blockDim.x + threadIdx.x;
    if (i < nE) atomicAdd(&deg[col[i]], 1.0f);
}

__global__ void rsqrt_kernel(float* __restrict__ d, int n) {
    int i = blockIdx.x * blockDim.x + threadIdx.x;
    if (i < n) d[i] = rsqrtf(d[i]);
}

// ---------------- fp32 WMMA GEMM: C[M,128] = A[M,128] @ W[128,128] ----------------
// One wave computes a 16x16 tile via V_WMMA_F32_16X16X4_F32, K-loop of 32 steps.
// Block = 256 threads = 8 waves -> covers 16 rows x all 128 cols. grid.x = M/16.
__global__ __launch_bounds__(256) void gemm_wmma_f32(const float* __restrict__ A,
                                                     const float* __restrict__ W,
                                                     float* __restrict__ C) {
    const int wave = threadIdx.x >> 5;
    const int lane = threadIdx.x & 31;
    const int half = lane >> 4;   // 0: lanes 0-15, 1: lanes 16-31
    const int l16  = lane & 15;
    const int m0 = blockIdx.x * 16;
    const int n0 = wave * 16;

    // A fragment source: row (m0+l16), this lane covers K = k + 2*half + {0,1}
    const float* arow = A + (size_t)(m0 + l16) * D + 2 * half;
    // B fragment source: column (n0+l16) of W, rows K = k + 2*half + {0,1}
    const float* wcol = W + n0 + l16 + (size_t)(2 * half) * D;

    v8f acc = {};
    #pragma unroll 4
    for (int k = 0; k < D; k += 4) {
        v2f a, b;
        a.x = arow[k];
        a.y = arow[k + 1];
        b.x = wcol[(size_t)k * D];
        b.y = wcol[(size_t)(k + 1) * D];
        acc = __builtin_amdgcn_wmma_f32_16x16x4_f32(
            /*neg_a=*/false, a, /*neg_b=*/false, b,
            /*c_mod=*/(short)0, acc, /*reuse_a=*/false, /*reuse_b=*/false);
    }

    // C/D layout: VGPR r, lane -> row = r + 8*half, col = l16
    float* crow = C + (size_t)(m0 + 8 * half) * D + n0 + l16;
    #pragma unroll
    for (int r = 0; r < 8; ++r) crow[(size_t)r * D] = acc[r];
}

// ---------------- edge aggregation: out[col] += h[row] * dinv[row]*dinv[col] ----------------
// One wave32 per edge: coalesced 512B float4 gather + 128 consecutive f32 atomics (L2-resident).
__global__ __launch_bounds__(256) void scatter_kernel(const float* __restrict__ h,
                                                      const int* __restrict__ row,
                                                      const int* __restrict__ col,
                                                      const float* __restrict__ dinv,
                                                      float* __restrict__ out,
                                                      int nE, int nN) {
    int gid  = blockIdx.x * blockDim.x + threadIdx.x;
    int e    = gid >> 5;
    int lane = gid & 31;
    if (e >= nE + nN) return;
    int r, c;
    if (e < nE) { r = row[e]; c = col[e]; }
    else        { r = c = e - nE; }           // self loop
    float nrm = dinv[r] * dinv[c];
    float4 v = ((const float4*)(h + (size_t)r * D))[lane];
    float* o = out + (size_t)c * D + lane * 4;
    atomicAdd(o + 0, v.x * nrm);
    atomicAdd(o + 1, v.y * nrm);
    atomicAdd(o + 2, v.z * nrm);
    atomicAdd(o + 3, v.w * nrm);
}

// ---------------- bias + relu (in place over N*128) ----------------
__global__ void bias_relu_kernel(float* __restrict__ h, const float* __restrict__ b, int total) {
    int i = blockIdx.x * blockDim.x + threadIdx.x;
    if (i < total) {
        float v = h[i] + b[i & (D - 1)];
        h[i] = v > 0.0f ? v : 0.0f;
    }
}

// ---------------- layer-3 projection: h3[n] = dot(h[n,:], W3) ----------------
__global__ __launch_bounds__(256) void dot_kernel(const float* __restrict__ h,
                                                  const float* __restrict__ w3,
                                                  float* __restrict__ out, int n) {
    int gid  = blockIdx.x * blockDim.x + threadIdx.x;
    int node = gid >> 5;
    int lane = gid & 31;
    if (node >= n) return;
    float4 a = ((const float4*)(h + (size_t)node * D))[lane];
    float4 b = ((const float4*)w3)[lane];
    float s = a.x * b.x + a.y * b.y + a.z * b.z + a.w * b.w;
    #pragma unroll
    for (int off = 16; off > 0; off >>= 1) s += __shfl_down(s, off, 32);
    if (lane == 0) out[node] = s;
}

// ---------------- layer-3 scalar scatter ----------------
__global__ void scatter1_kernel(const float* __restrict__ h3,
                                const int* __restrict__ row, const int* __restrict__ col,
                                const float* __restrict__ dinv, float* __restrict__ out,
                                int nE, int nN) {
    int e = blockIdx.x * blockDim.x + threadIdx.x;
    if (e >= nE + nN) return;
    int r, c;
    if (e < nE) { r = row[e]; c = col[e]; }
    else        { r = c = e - nE; }
    atomicAdd(&out[c], h3[r] * dinv[r] * dinv[c]);
}

// ---------------- sigmoid(agg + b3) ----------------
__global__ void sigmoid_kernel(const float* __restrict__ agg, const float* __restrict__ b3,
                               float* __restrict__ out, int n) {
    int i = blockIdx.x * blockDim.x + threadIdx.x;
    if (i < n) {
        float v = agg[i] + b3[0];
        out[i] = 1.0f / (1.0f + __expf(-v));
    }
}

extern "C" void kernel_launch(void* const* d_in, const int* in_sizes, int n_in,
                              void* d_out, int out_size, void* d_ws, size_t ws_size,
                              hipStream_t stream) {
    const float* x   = (const float*)d_in[0];
    const int*   ei  = (const int*)d_in[1];
    const float* W1  = (const float*)d_in[2];
    const float* b1  = (const float*)d_in[3];
    const float* W2  = (const float*)d_in[4];
    const float* b2  = (const float*)d_in[5];
    const float* W3  = (const float*)d_in[6];
    const float* b3  = (const float*)d_in[7];

    const int N = in_sizes[0] / D;      // 100000
    const int E = in_sizes[1] / 2;      // 3200000
    const int* rowp = ei;               // sources
    const int* colp = ei + E;           // destinations

    // workspace layout
    char*  ws   = (char*)d_ws;
    float* deg  = (float*)ws;             // N (becomes dinv)
    float* h3   = deg + N;                // N
    float* agg3 = h3 + N;                 // N
    size_t off  = ((size_t)3 * N * sizeof(float) + 255) & ~(size_t)255;
    float* bufA = (float*)(ws + off);                 // N*128 (GEMM out)
    float* bufB = bufA + (size_t)N * D;               // N*128 (agg / activ)

    const int BT = 256;
    const int total   = N * D;
    const int nblkF4  = (total / 4 + BT - 1) / BT;
    const int nblkN   = (N + BT - 1) / BT;
    const int nblkE   = (E + BT - 1) / BT;
    const int nblkEN  = (E + N + BT - 1) / BT;
    const int nblkSc  = (int)(((long long)(E + N) * 32 + BT - 1) / BT);
    const int nblkDot = (int)(((long long)N * 32 + BT - 1) / BT);
    const int nblkG   = N / 16;          // 100000/16 = 6250 exactly

    // normalization: deg = 1 (self loop) + indegree; dinv = rsqrt(deg)
    fill_kernel<<<nblkN, BT, 0, stream>>>(deg, 1.0f, N);
    degree_kernel<<<nblkE, BT, 0, stream>>>(colp, deg, E);
    rsqrt_kernel<<<nblkN, BT, 0, stream>>>(deg, N);

    // ---- layer 1 ----
    gemm_wmma_f32<<<nblkG, BT, 0, stream>>>(x, W1, bufA);
    fill4_kernel<<<nblkF4, BT, 0, stream>>>((float4*)bufB, 0.0f, total / 4);
    scatter_kernel<<<nblkSc, BT, 0, stream>>>(bufA, rowp, colp, deg, bufB, E, N);
    bias_relu_kernel<<<(total + BT - 1) / BT, BT, 0, stream>>>(bufB, b1, total);

    // ---- layer 2 ----
    gemm_wmma_f32<<<nblkG, BT, 0, stream>>>(bufB, W2, bufA);
    fill4_kernel<<<nblkF4, BT, 0, stream>>>((float4*)bufB, 0.0f, total / 4);
    scatter_kernel<<<nblkSc, BT, 0, stream>>>(bufA, rowp, colp, deg, bufB, E, N);
    bias_relu_kernel<<<(total + BT - 1) / BT, BT, 0, stream>>>(bufB, b2, total);

    // ---- layer 3 (HID -> 1) ----
    dot_kernel<<<nblkDot, BT, 0, stream>>>(bufB, W3, h3, N);
    fill_kernel<<<nblkN, BT, 0, stream>>>(agg3, 0.0f, N);
    scatter1_kernel<<<nblkEN, BT, 0, stream>>>(h3, rowp, colp, deg, agg3, E, N);
    sigmoid_kernel<<<nblkN, BT, 0, stream>>>(agg3, b3, (float*)d_out, N);
}